// GCN_24043226923838
// MI455X (gfx1250) — compile-verified
//
#include <hip/hip_runtime.h>
#include <hip/hip_bf16.h>

typedef __attribute__((ext_vector_type(2))) float v2f;
typedef __attribute__((ext_vector_type(8))) float v8f;
typedef __attribute__((ext_vector_type(4))) int   iv4;

// ds_swizzle XOR-mode immediates (group-of-32): imm = xor<<10 | or<<5 | and(0x1f)
#define SWZ_X1  0x041f
#define SWZ_X2  0x081f
#define SWZ_X4  0x101f
#define SWZ_X8  0x201f
#define SWZ_X16 0x401f

__device__ __forceinline__ float swz_add(float v, const int imm_shifted_is_constexpr) {
    return v; // placeholder (unused)
}

// ---------------------------------------------------------------------------
// K1: degree accumulation. Index streams (128MB) use non-temporal b128 loads
// so they don't evict the L2-resident node arrays. 4 edges per thread/iter.
// ---------------------------------------------------------------------------
__global__ void gcn_degrees(const int* __restrict__ src, const int* __restrict__ dst,
                            float* __restrict__ deg_out, float* __restrict__ deg_in,
                            int n_edges) {
    int tid    = blockIdx.x * blockDim.x + threadIdx.x;
    int stride = gridDim.x * blockDim.x;
    int n4     = n_edges >> 2;
    const iv4* s4 = (const iv4*)src;
    const iv4* d4 = (const iv4*)dst;
    for (int e = tid; e < n4; e += stride) {
        iv4 s = __builtin_nontemporal_load(&s4[e]);
        iv4 d = __builtin_nontemporal_load(&d4[e]);
        atomicAdd(&deg_out[s.x], 1.0f); atomicAdd(&deg_in[d.x], 1.0f);
        atomicAdd(&deg_out[s.y], 1.0f); atomicAdd(&deg_in[d.y], 1.0f);
        atomicAdd(&deg_out[s.z], 1.0f); atomicAdd(&deg_in[d.z], 1.0f);
        atomicAdd(&deg_out[s.w], 1.0f); atomicAdd(&deg_in[d.w], 1.0f);
    }
    for (int e = n4 * 4 + tid; e < n_edges; e += stride) {          // tail
        atomicAdd(&deg_out[src[e]], 1.0f);
        atomicAdd(&deg_in[dst[e]], 1.0f);
    }
}

// ---------------------------------------------------------------------------
// K2: xs[n] = in_feat[n] * rsqrt(max(deg_out[n],1))   (layer-1 out-norm)
// ---------------------------------------------------------------------------
__global__ void gcn_scale_in(const float* __restrict__ in_feat,
                             const float* __restrict__ deg_out,
                             float* __restrict__ xs, int n_nodes) {
    int i = blockIdx.x * blockDim.x + threadIdx.x;
    if (i < n_nodes) {
        float s = rsqrtf(fmaxf(deg_out[i], 1.0f));
        float4 v = ((const float4*)in_feat)[i];
        v.x *= s; v.y *= s; v.z *= s; v.w *= s;
        ((float4*)xs)[i] = v;
    }
}

// ---------------------------------------------------------------------------
// K3: layer-1 message passing: macc[dst] += xs[src]  (4 features/edge)
// ---------------------------------------------------------------------------
__global__ void gcn_scatter4(const int* __restrict__ src, const int* __restrict__ dst,
                             const float* __restrict__ xs, float* __restrict__ macc,
                             int n_edges) {
    int tid    = blockIdx.x * blockDim.x + threadIdx.x;
    int stride = gridDim.x * blockDim.x;
    int n4     = n_edges >> 2;
    const iv4* s4 = (const iv4*)src;
    const iv4* d4 = (const iv4*)dst;
    for (int e = tid; e < n4; e += stride) {
        iv4 s = __builtin_nontemporal_load(&s4[e]);
        iv4 d = __builtin_nontemporal_load(&d4[e]);
#pragma unroll
        for (int k = 0; k < 4; ++k) {
            int si = (k == 0) ? s.x : (k == 1) ? s.y : (k == 2) ? s.z : s.w;
            int di = (k == 0) ? d.x : (k == 1) ? d.y : (k == 2) ? d.z : d.w;
            float4 v = ((const float4*)xs)[si];            // L2-resident gather
            float* p = &macc[(size_t)di * 4];
            atomicAdd(p + 0, v.x);
            atomicAdd(p + 1, v.y);
            atomicAdd(p + 2, v.z);
            atomicAdd(p + 3, v.w);
        }
    }
    for (int e = n4 * 4 + tid; e < n_edges; e += stride) {          // tail
        int si = src[e], di = dst[e];
        float4 v = ((const float4*)xs)[si];
        float* p = &macc[(size_t)di * 4];
        atomicAdd(p + 0, v.x); atomicAdd(p + 1, v.y);
        atomicAdd(p + 2, v.z); atomicAdd(p + 3, v.w);
    }
}

// ---------------------------------------------------------------------------
// K4: fused node update via WMMA f32 16x16x4.
//   A (16x4)  = macc[16 nodes] * norm_in     (ISA 7.12.2 f32 A layout)
//   B (4x16)  = W1
//   D = A@B ; v[m,n] = relu(D+b1[n]) * W2[n] ; row-sum via ds_swizzle
//   butterflies (interleaved across the 8 rows for ILP); norm_out applied
//   once per node after the reduction; single coalesced guarded store.
// ---------------------------------------------------------------------------
__global__ void gcn_layer1_wmma(const float* __restrict__ macc,
                                const float* __restrict__ deg_in,
                                const float* __restrict__ deg_out,
                                const float* __restrict__ W1,   // [4][16] row-major
                                const float* __restrict__ b1,   // [16]
                                const float* __restrict__ W2,   // [16]
                                float* __restrict__ t, int n_nodes) {
    int gwave = (blockIdx.x * blockDim.x + threadIdx.x) >> 5;
    int lane  = threadIdx.x & 31;
    int base  = gwave * 16;
    if (base >= n_nodes) return;          // wave-uniform: EXEC stays all-1s

    int half = lane >> 4;                 // 0: lanes 0-15, 1: lanes 16-31
    int mn   = lane & 15;                 // row index for A, col index for B/D

    // A: lane holds K={2*half, 2*half+1} of row mn
    int nodeA = min(base + mn, n_nodes - 1);
    float nin = rsqrtf(fmaxf(deg_in[nodeA], 1.0f));
    const float* arow = &macc[(size_t)nodeA * 4 + half * 2];
    v2f a; a.x = arow[0] * nin; a.y = arow[1] * nin;

    // B: lane holds rows K={2*half, 2*half+1} at column mn
    v2f b; b.x = W1[(half * 2) * 16 + mn]; b.y = W1[(half * 2 + 1) * 16 + mn];

    v8f c = {};
    c = __builtin_amdgcn_wmma_f32_16x16x4_f32(
        false, a, false, b, (short)0, c, false, false);

    float bn = b1[mn];
    float w2 = W2[mn];

    float v[8];
#pragma unroll
    for (int r = 0; r < 8; ++r) v[r] = fmaxf(c[r] + bn, 0.0f) * w2;

    // 16-lane XOR butterfly, one level at a time across all 8 rows (ILP)
#pragma unroll
    for (int r = 0; r < 8; ++r)
        v[r] += __int_as_float(__builtin_amdgcn_ds_swizzle(__float_as_int(v[r]), SWZ_X1));
#pragma unroll
    for (int r = 0; r < 8; ++r)
        v[r] += __int_as_float(__builtin_amdgcn_ds_swizzle(__float_as_int(v[r]), SWZ_X2));
#pragma unroll
    for (int r = 0; r < 8; ++r)
        v[r] += __int_as_float(__builtin_amdgcn_ds_swizzle(__float_as_int(v[r]), SWZ_X4));
#pragma unroll
    for (int r = 0; r < 8; ++r)
        v[r] += __int_as_float(__builtin_amdgcn_ds_swizzle(__float_as_int(v[r]), SWZ_X8));

    // lane mn==r keeps row r's sum; rows live on lanes 0-7 (half 0) / 16-23 (half 1)
    float res = v[0];
#pragma unroll
    for (int r = 1; r < 8; ++r) res = (mn == r) ? v[r] : res;

    int node = base + mn + 8 * half;
    if (mn < 8 && node < n_nodes) {
        float nout = rsqrtf(fmaxf(deg_out[node], 1.0f));   // layer-2 pre-scale
        t[node] = res * nout;
    }
}

// ---------------------------------------------------------------------------
// K5: layer-2 message passing: m2[dst] += t[src]  (1 float/edge)
// ---------------------------------------------------------------------------
__global__ void gcn_scatter1(const int* __restrict__ src, const int* __restrict__ dst,
                             const float* __restrict__ t, float* __restrict__ m2,
                             int n_edges) {
    int tid    = blockIdx.x * blockDim.x + threadIdx.x;
    int stride = gridDim.x * blockDim.x;
    int n4     = n_edges >> 2;
    const iv4* s4 = (const iv4*)src;
    const iv4* d4 = (const iv4*)dst;
    for (int e = tid; e < n4; e += stride) {
        iv4 s = __builtin_nontemporal_load(&s4[e]);
        iv4 d = __builtin_nontemporal_load(&d4[e]);
        atomicAdd(&m2[d.x], t[s.x]);
        atomicAdd(&m2[d.y], t[s.y]);
        atomicAdd(&m2[d.z], t[s.z]);
        atomicAdd(&m2[d.w], t[s.w]);
    }
    for (int e = n4 * 4 + tid; e < n_edges; e += stride) {          // tail
        atomicAdd(&m2[dst[e]], t[src[e]]);
    }
}

// ---------------------------------------------------------------------------
// K6: h2[n] = m2[n]*norm_in + b2; pool into per-graph sum/count.
// node2graph is sorted -> waves are usually graph-uniform: collapse 32
// same-address atomics into 1 via a wave swizzle reduction.
// ---------------------------------------------------------------------------
__global__ void gcn_pool(const float* __restrict__ m2, const float* __restrict__ deg_in,
                         const float* __restrict__ b2, const int* __restrict__ node2graph,
                         float* __restrict__ gsum, float* __restrict__ gcnt, int n_nodes) {
    int i    = blockIdx.x * blockDim.x + threadIdx.x;
    int lane = threadIdx.x & 31;
    bool valid = i < n_nodes;
    int   g  = valid ? node2graph[i] : -1;
    float h2 = 0.0f;
    if (valid) h2 = m2[i] * rsqrtf(fmaxf(deg_in[i], 1.0f)) + b2[0];

    int g0 = __shfl(g, 0);
    if (__all(g == g0)) {
        if (g0 < 0) return;                 // entire wave out of range
        h2 += __int_as_float(__builtin_amdgcn_ds_swizzle(__float_as_int(h2), SWZ_X1));
        h2 += __int_as_float(__builtin_amdgcn_ds_swizzle(__float_as_int(h2), SWZ_X2));
        h2 += __int_as_float(__builtin_amdgcn_ds_swizzle(__float_as_int(h2), SWZ_X4));
        h2 += __int_as_float(__builtin_amdgcn_ds_swizzle(__float_as_int(h2), SWZ_X8));
        h2 += __int_as_float(__builtin_amdgcn_ds_swizzle(__float_as_int(h2), SWZ_X16));
        if (lane == 0) {
            atomicAdd(&gsum[g0], h2);
            atomicAdd(&gcnt[g0], 32.0f);    // uniform wave => all 32 lanes valid
        }
    } else if (valid) {
        atomicAdd(&gsum[g], h2);
        atomicAdd(&gcnt[g], 1.0f);
    }
}

// ---------------------------------------------------------------------------
// K7: out[g] = gsum[g] / max(gcnt[g], 1)
// ---------------------------------------------------------------------------
__global__ void gcn_finalize(const float* __restrict__ gsum, const float* __restrict__ gcnt,
                             float* __restrict__ out, int n_graphs) {
    int i = blockIdx.x * blockDim.x + threadIdx.x;
    if (i < n_graphs) out[i] = gsum[i] / fmaxf(gcnt[i], 1.0f);
}

// ---------------------------------------------------------------------------
static inline size_t align256(size_t x) { return (x + 255) & ~(size_t)255; }

extern "C" void kernel_launch(void* const* d_in, const int* in_sizes, int n_in,
                              void* d_out, int out_size, void* d_ws, size_t ws_size,
                              hipStream_t stream) {
    const float* in_feat    = (const float*)d_in[0];   // [N,4]
    const int*   src        = (const int*)d_in[1];     // [E]
    const int*   dst        = (const int*)d_in[2];     // [E]
    const int*   node2graph = (const int*)d_in[3];     // [N]
    const float* W1         = (const float*)d_in[4];   // [4,16]
    const float* b1         = (const float*)d_in[5];   // [16]
    const float* W2         = (const float*)d_in[6];   // [16,1]
    const float* b2         = (const float*)d_in[7];   // [1]
    float*       out        = (float*)d_out;           // [G,1]

    const int n_nodes  = in_sizes[0] / 4;
    const int n_edges  = in_sizes[1];
    const int n_graphs = out_size;

    // Workspace layout (~24 MB)
    char* ws = (char*)d_ws;
    size_t off = 0;
    float* deg_out = (float*)(ws + off); off = align256(off + (size_t)n_nodes * 4);
    float* deg_in  = (float*)(ws + off); off = align256(off + (size_t)n_nodes * 4);
    float* xs      = (float*)(ws + off); off = align256(off + (size_t)n_nodes * 16);
    float* macc    = (float*)(ws + off); off = align256(off + (size_t)n_nodes * 16);
    float* tbuf    = (float*)(ws + off); off = align256(off + (size_t)n_nodes * 4);
    float* m2      = (float*)(ws + off); off = align256(off + (size_t)n_nodes * 4);
    float* gsum    = (float*)(ws + off); off = align256(off + (size_t)n_graphs * 4);
    float* gcnt    = (float*)(ws + off); off = align256(off + (size_t)n_graphs * 4);
    size_t total   = off;
    (void)ws_size;

    hipMemsetAsync(d_ws, 0, total, stream);            // capture-safe zeroing

    const int TB = 256;
    const int edge_blocks = 2048;                      // grid-stride, 4 edges/thread/iter
    const int node_blocks = (n_nodes + TB - 1) / TB;

    gcn_degrees<<<edge_blocks, TB, 0, stream>>>(src, dst, deg_out, deg_in, n_edges);
    gcn_scale_in<<<node_blocks, TB, 0, stream>>>(in_feat, deg_out, xs, n_nodes);
    gcn_scatter4<<<edge_blocks, TB, 0, stream>>>(src, dst, xs, macc, n_edges);

    int n_waves  = (n_nodes + 15) / 16;
    int w_blocks = (n_waves * 32 + TB - 1) / TB;
    gcn_layer1_wmma<<<w_blocks, TB, 0, stream>>>(macc, deg_in, deg_out,
                                                 W1, b1, W2, tbuf, n_nodes);

    gcn_scatter1<<<edge_blocks, TB, 0, stream>>>(src, dst, tbuf, m2, n_edges);
    gcn_pool<<<node_blocks, TB, 0, stream>>>(m2, deg_in, b2, node2graph,
                                             gsum, gcnt, n_nodes);
    int g_blocks = (n_graphs + TB - 1) / TB;
    gcn_finalize<<<g_blocks, TB, 0, stream>>>(gsum, gcnt, out, n_graphs);
}